// Net_2525440770051
// MI455X (gfx1250) — compile-verified
//
#include <hip/hip_runtime.h>
#include <hip/hip_bf16.h>

typedef __attribute__((ext_vector_type(2))) float v2f;
typedef __attribute__((ext_vector_type(8))) float v8f;

// ---------------------------------------------------------------------------
// Conv1d-as-GEMM with relu, fp32 WMMA (V_WMMA_F32_16X16X4_F32).
// Fully template-specialized per layer: im2col index math and tile
// decomposition are compile-time (TM = Cout/16 is a power of two).
//   in  : [CIN][Tin]   channel-major activations
//   w   : [Cout][CIN*KW] (torch OIH layout flattens to exactly this)
//   out : [Cout][Tout] channel-major
// One wave32 owns a 16(M) x 64(N) strip = NT(4) 16x16 WMMA tiles sharing one
// A (weight) fragment per K-step:  1 A-load -> 4 v_wmma.
// A-matrix lane layout (ISA 7.12.2, 32-bit A 16x4):
//   lane L: M = L&15;  VGPR0 = A[M][2*(L>>4)], VGPR1 = A[M][2*(L>>4)+1]
// B-matrix (4x16, rows striped across VGPRs like C/D):
//   lane L: N = L&15;  VGPR0 = B[2*(L>>4)][N], VGPR1 = B[2*(L>>4)+1][N]
// C/D: lane L: N = L&15; VGPR v = D[v + 8*(L>>4)][N]
// ---------------------------------------------------------------------------
template <int CIN, int KW, int STRIDE, int PAD, int TM>
__global__ __launch_bounds__(256) void conv_wmma_kernel(
    const float* __restrict__ in,
    const float* __restrict__ w,
    const float* __restrict__ bias,
    float* __restrict__ out,
    int Tin, int Tout, int n_groups)
{
    constexpr int KTOT = CIN * KW;
    constexpr int NT   = 4;            // 16x16 N-tiles per wave

    const int wid = blockIdx.x * (blockDim.x >> 5) + (threadIdx.x >> 5);
    if (wid >= TM * n_groups) return;  // wave-uniform exit: EXEC stays all-1s

    const int tile_m = wid & (TM - 1);     // TM is a power of two
    const int grp_n  = wid / TM;           // folds to a shift

    const int lane = threadIdx.x & 31;
    const int half = lane >> 4;            // 0: lanes 0-15, 1: lanes 16-31
    const int lo   = lane & 15;

    const int m = tile_m * 16 + lo;        // A row: output channel

    // Per-subtile output time index; clamp out-of-range columns instead of
    // predicating every load: the conv arithmetic guarantees
    // (Tout-1)*STRIDE + KW-1 <= Tin-1+2*PAD, so clamped lanes read in-bounds
    // (PAD==0) and are simply never stored.
    int t[NT], tbase[NT];
#pragma unroll
    for (int i = 0; i < NT; ++i) {
        t[i] = (grp_n * NT + i) * 16 + lo;
        const int tc = t[i] < Tout ? t[i] : (Tout - 1);
        tbase[i] = tc * STRIDE - PAD;
    }

    const float* __restrict__ wrow = w + (size_t)m * (size_t)KTOT;

    v8f acc[NT];
#pragma unroll
    for (int i = 0; i < NT; ++i) acc[i] = {};

#pragma unroll 4
    for (int kb = 0; kb < KTOT; kb += 4) {
        const int r0 = kb + 2 * half;

        // Shared A fragment: two consecutive weights of row m (one b64 load)
        v2f a;
        a.x = wrow[r0];
        a.y = wrow[r0 + 1];

        // im2col row r = c*KW + j  ->  in[c][t*STRIDE + j - PAD]
        const int c0 = r0 / KW;
        const int j0 = r0 % KW;
        const int c1 = (r0 + 1) / KW;
        const int j1 = (r0 + 1) % KW;
        const size_t off0 = (size_t)c0 * Tin + j0;   // common across subtiles
        const size_t off1 = (size_t)c1 * Tin + j1;

#pragma unroll
        for (int i = 0; i < NT; ++i) {
            v2f b;
            if constexpr (PAD == 0) {
                b.x = in[off0 + tbase[i]];
                b.y = in[off1 + tbase[i]];
            } else {
                const int it0 = tbase[i] + j0;
                const int it1 = tbase[i] + j1;
                b.x = (it0 >= 0 && it0 < Tin) ? in[off0 + tbase[i]] : 0.0f;
                b.y = (it1 >= 0 && it1 < Tin) ? in[off1 + tbase[i]] : 0.0f;
            }
            // D = A(16x4,f32) * B(4x16,f32) + C  -- full fp32 precision
            acc[i] = __builtin_amdgcn_wmma_f32_16x16x4_f32(
                false, a, false, b, (short)0, acc[i], false, false);
        }
    }

#pragma unroll
    for (int i = 0; i < NT; ++i) {
        if (t[i] < Tout) {
#pragma unroll
            for (int v = 0; v < 8; ++v) {
                const int oc = tile_m * 16 + v + 8 * half;
                const float val = acc[i][v] + bias[oc];
                out[(size_t)oc * Tout + t[i]] = fmaxf(val, 0.0f);
            }
        }
    }
}

// ---------------------------------------------------------------------------
// gi[t][g] = sum_c act[c][t] * w_ih[g][c] + b_ih[g],  g in {r,z,n}
// act reads are coalesced over t; w_ih reads are wave-uniform (scalar loads).
// ---------------------------------------------------------------------------
__global__ __launch_bounds__(256) void gi_kernel(
    const float* __restrict__ act,   // [256][T]
    const float* __restrict__ w_ih,  // [3][256]
    const float* __restrict__ b_ih,  // [3]
    float* __restrict__ gi,          // [T][3]
    int T)
{
    const int t = blockIdx.x * blockDim.x + threadIdx.x;
    if (t >= T) return;
    float g0 = b_ih[0], g1 = b_ih[1], g2 = b_ih[2];
#pragma unroll 4
    for (int c = 0; c < 256; ++c) {
        const float a = act[(size_t)c * T + t];
        g0 = fmaf(a, w_ih[c],        g0);
        g1 = fmaf(a, w_ih[256 + c],  g1);
        g2 = fmaf(a, w_ih[512 + c],  g2);
    }
    gi[t * 3 + 0] = g0;
    gi[t * 3 + 1] = g1;
    gi[t * 3 + 2] = g2;
}

// ---------------------------------------------------------------------------
// Scalar-hidden GRU scan: nonlinear recurrence, inherently sequential.
// One lane, minimal per-step latency (fmaf + fast exp).
// Emits the last 8 hidden states -> d_out[0..7].
// ---------------------------------------------------------------------------
__global__ void gru_kernel(
    const float* __restrict__ gi,    // [T][3]
    const float* __restrict__ w_hh,  // [3][1]
    const float* __restrict__ b_hh,  // [3]
    const float* __restrict__ hx,    // [1][1]
    float* __restrict__ out,         // [8]
    int T)
{
    if (threadIdx.x != 0 || blockIdx.x != 0) return;
    float h = hx[0];
    const float whr = w_hh[0], whz = w_hh[1], whn = w_hh[2];
    const float bhr = b_hh[0], bhz = b_hh[1], bhn = b_hh[2];
    for (int t = 0; t < T; ++t) {
        const float gir = gi[t * 3 + 0];
        const float giz = gi[t * 3 + 1];
        const float gin = gi[t * 3 + 2];
        const float ghr = fmaf(h, whr, bhr);
        const float ghz = fmaf(h, whz, bhz);
        const float ghn = fmaf(h, whn, bhn);
        const float r = 1.0f / (1.0f + __expf(-(gir + ghr)));
        const float z = 1.0f / (1.0f + __expf(-(giz + ghz)));
        const float n = tanhf(fmaf(r, ghn, gin));
        h = (1.0f - z) * n + z * h;
        if (t >= T - 8) out[t - (T - 8)] = h;
    }
}

// ---------------------------------------------------------------------------
extern "C" void kernel_launch(void* const* d_in, const int* in_sizes, int n_in,
                              void* d_out, int out_size, void* d_ws, size_t ws_size,
                              hipStream_t stream) {
    const float* x    = (const float*)d_in[0];
    const float* w1   = (const float*)d_in[1];
    const float* b1   = (const float*)d_in[2];
    const float* w2   = (const float*)d_in[3];
    const float* b2   = (const float*)d_in[4];
    const float* w3   = (const float*)d_in[5];
    const float* b3   = (const float*)d_in[6];
    const float* w4   = (const float*)d_in[7];
    const float* b4   = (const float*)d_in[8];
    const float* w5   = (const float*)d_in[9];
    const float* b5   = (const float*)d_in[10];
    const float* w_ih = (const float*)d_in[11];
    const float* b_ih = (const float*)d_in[12];
    const float* w_hh = (const float*)d_in[13];
    const float* b_hh = (const float*)d_in[14];
    const float* hx   = (const float*)d_in[15];
    float* out = (float*)d_out;

    const int L  = in_sizes[0];              // 16,777,216
    const int L1 = (L  - 32) / 16 + 1;       // 1,048,575
    const int L2 = (L1 - 16) / 8  + 1;       // 131,070
    const int L3 = (L2 - 16) / 8  + 1;       // 16,382
    const int L4 = (L3 - 4)  / 2  + 1;       // 8,190
    const int L5 = L4;                       // 8,190 (pad=1, k=3, stride=1)

    // Workspace ping-pong: bufA holds act1 -> act3 -> act5, bufB holds act2 -> act4.
    float* bufA = (float*)d_ws;
    float* bufB = bufA + (size_t)32 * (size_t)L1;
    float* gi   = bufB + (size_t)64 * (size_t)L2;

    // waves = TM * n_groups; n_groups = ceil(ceil(Tout/16)/4); 8 waves/block
    auto blocks_for = [](int TM, int Tout, int& ng) {
        const int tn = (Tout + 15) / 16;
        ng = (tn + 3) / 4;
        return (TM * ng + 7) / 8;
    };

    int ng, nb;

    nb = blocks_for(2, L1, ng);
    conv_wmma_kernel<1, 32, 16, 0, 2><<<nb, 256, 0, stream>>>(
        x, w1, b1, bufA, L, L1, ng);                        // act1 [32][L1]

    nb = blocks_for(4, L2, ng);
    conv_wmma_kernel<32, 16, 8, 0, 4><<<nb, 256, 0, stream>>>(
        bufA, w2, b2, bufB, L1, L2, ng);                    // act2 [64][L2]

    nb = blocks_for(8, L3, ng);
    conv_wmma_kernel<64, 16, 8, 0, 8><<<nb, 256, 0, stream>>>(
        bufB, w3, b3, bufA, L2, L3, ng);                    // act3 [128][L3]

    nb = blocks_for(16, L4, ng);
    conv_wmma_kernel<128, 4, 2, 0, 16><<<nb, 256, 0, stream>>>(
        bufA, w4, b4, bufB, L3, L4, ng);                    // act4 [256][L4]

    nb = blocks_for(16, L5, ng);
    conv_wmma_kernel<256, 3, 1, 1, 16><<<nb, 256, 0, stream>>>(
        bufB, w5, b5, bufA, L4, L5, ng);                    // act5 [256][L5]

    gi_kernel<<<(L5 + 255) / 256, 256, 0, stream>>>(bufA, w_ih, b_ih, gi, L5);
    gru_kernel<<<1, 32, 0, stream>>>(gi, w_hh, b_hh, hx, out, L5);
}